// Cnn_5626407158438
// MI455X (gfx1250) — compile-verified
//
#include <hip/hip_runtime.h>
#include <hip/hip_bf16.h>

typedef __attribute__((ext_vector_type(2))) float v2f;
typedef __attribute__((ext_vector_type(8))) float v8f;

#define NPTS 10000
#define NPARA 5000
#define WAVES_PER_WG 4
#define PTS_PER_WAVE 16
#define PTS_PER_WG (WAVES_PER_WG * PTS_PER_WAVE)

// ---- LDS layout (float offsets) ----
#define OFF_W     0              // Wpad[4][64][52], zero-padded from 50x50
#define W_LSTRIDE (64 * 52)      // 3328
#define OFF_B     13312          // bpad[4][64]
#define OFF_W1    13568          // W1 pad [64][2]
#define OFF_B1    13696          // b1 pad [64]
#define OFF_W6    13760          // W6 pad [64]
#define OFF_H     13824          // per-wave ping-pong stream panels
// H panel is column-major per point: element (k,p) at p*P_STRIDE + k.
// Stride 68: lane p touches dword-banks 4p+k -> conflict-free full-wave b64 loads.
#define P_STRIDE  68
#define H_BUF     (16 * P_STRIDE)        // 1088 floats per stream panel
#define H_WAVE    (2 * 5 * H_BUF)        // ping-pong x 5 streams = 10880
#define SMEM_FLOATS (OFF_H + WAVES_PER_WG * H_WAVE)   // 57344 floats = 229376 B (< 320KB WGP LDS)

__global__ void zero_acc_kernel(float* acc) {
    if (threadIdx.x < 10) acc[threadIdx.x] = 0.0f;
}

__global__ __launch_bounds__(WAVES_PER_WG * 32)
void pinn_point_kernel(const float* __restrict__ x,
                       const float* __restrict__ W1, const float* __restrict__ b1,
                       const float* __restrict__ W2, const float* __restrict__ b2,
                       const float* __restrict__ W3, const float* __restrict__ b3,
                       const float* __restrict__ W4, const float* __restrict__ b4,
                       const float* __restrict__ W5, const float* __restrict__ b5,
                       const float* __restrict__ W6, const float* __restrict__ b6,
                       float* __restrict__ acc)
{
    extern __shared__ float sm[];
    const int tid = threadIdx.x;
    const int nthr = blockDim.x;

    // ---- stage weights into LDS, zero-padded (pad rows/cols make padded WMMA tiles exact) ----
    {
        const float* Wl[4] = {W2, W3, W4, W5};
        for (int l = 0; l < 4; ++l) {
            const float* W = Wl[l];
            for (int i = tid; i < 64 * 52; i += nthr) {
                int r = i / 52, c = i % 52;
                sm[OFF_W + l * W_LSTRIDE + i] = (r < 50 && c < 50) ? W[r * 50 + c] : 0.0f;
            }
        }
        const float* bl[4] = {b2, b3, b4, b5};
        for (int l = 0; l < 4; ++l)
            for (int i = tid; i < 64; i += nthr)
                sm[OFF_B + l * 64 + i] = (i < 50) ? bl[l][i] : 0.0f;
        for (int i = tid; i < 64; i += nthr) {
            sm[OFF_W1 + 2 * i]     = (i < 50) ? W1[2 * i]     : 0.0f;
            sm[OFF_W1 + 2 * i + 1] = (i < 50) ? W1[2 * i + 1] : 0.0f;
            sm[OFF_B1 + i] = (i < 50) ? b1[i] : 0.0f;
            sm[OFF_W6 + i] = (i < 50) ? W6[i] : 0.0f;
        }
    }
    __syncthreads();

    const int wave = tid >> 5;
    const int lane = tid & 31;
    const int p  = lane & 15;     // point within the wave's 16-point group
    const int hi = lane >> 4;     // half-wave selector used by WMMA fragment layouts
    float* H = sm + OFF_H + wave * H_WAVE;

    const int gp = blockIdx.x * PTS_PER_WG + wave * PTS_PER_WAVE + p;
    const bool valid = (gp < NPTS);
    const float xp = valid ? x[2 * gp]     : 0.0f;
    const float tp = valid ? x[2 * gp + 1] : 0.0f;

    // ---- layer 1 (K=2, trivial) + tanh-Taylor, written into buffer q=0 ----
    // streams: 0=value, 1=d/dx, 2=d2/dx2, 3=d3/dx3, 4=d/dt
    for (int i = 0; i < 32; ++i) {
        const int r = hi + 2 * i;                 // each lane covers 32 of the 64 padded rows
        const float w0 = sm[OFF_W1 + 2 * r];
        const float w1 = sm[OFF_W1 + 2 * r + 1];
        const float zv = w0 * xp + w1 * tp + sm[OFF_B1 + r];
        const float a  = tanhf(zv);
        const float s  = 1.0f - a * a;
        const float ax   = s * w0;
        const float axx  = -2.0f * a * w0 * ax;
        const float axxx = (4.0f * a * a - 2.0f * s) * w0 * w0 * ax;
        const float at   = s * w1;
        const int idx = p * P_STRIDE + r;
        H[0 * H_BUF + idx] = a;
        H[1 * H_BUF + idx] = ax;
        H[2 * H_BUF + idx] = axx;
        H[3 * H_BUF + idx] = axxx;
        H[4 * H_BUF + idx] = at;
    }
    // No barrier: each wave's H panels are private; LDS ops are in-order per wave.

    // ---- hidden layers 2..5: Z = W @ H via f32 WMMA 16x16x4, then tanh-Taylor ----
    int q = 0;
    for (int l = 0; l < 4; ++l) {
        const float* Wl_ = sm + OFF_W + l * W_LSTRIDE;
        const float* bb  = sm + OFF_B + l * 64;
        const float* Hin = H + q * 5 * H_BUF;
        float* Hout      = H + (q ^ 1) * 5 * H_BUF;

        for (int m = 0; m < 4; ++m) {            // 4 row-tiles cover 64 (padded) output rows
            v8f C[5] = {};
            const int arow = m * 16 + p;         // A-fragment row for this lane
            for (int ks = 0; ks < 13; ++ks) {    // K = 52 (padded from 50), 4 per step
                const int k0 = 4 * ks + 2 * hi;  // even -> 8B-aligned b64 loads
                const v2f A = *(const v2f*)&Wl_[arow * 52 + k0];
#pragma unroll
                for (int d = 0; d < 5; ++d) {
                    const v2f B = *(const v2f*)&Hin[d * H_BUF + p * P_STRIDE + k0];
                    C[d] = __builtin_amdgcn_wmma_f32_16x16x4_f32(
                        false, A, false, B, (short)0, C[d], false, false);
                }
            }
#pragma unroll
            for (int j = 0; j < 8; ++j) {
                const int row = m * 16 + j + 8 * hi;   // C layout: VGPR j -> row j (+8 for hi half)
                const float zv   = C[0][j] + bb[row];
                const float a    = tanhf(zv);
                const float s    = 1.0f - a * a;
                const float zx   = C[1][j];
                const float zxx  = C[2][j];
                const float zxxx = C[3][j];
                const float zt   = C[4][j];
                const float ax   = s * zx;
                const float axx  = s * zxx - 2.0f * a * zx * ax;
                const float axxx = s * zxxx - 6.0f * a * ax * zxx
                                 + (4.0f * a * a - 2.0f * s) * zx * zx * ax;
                const float at   = s * zt;
                const int idx = p * P_STRIDE + row;
                Hout[0 * H_BUF + idx] = a;
                Hout[1 * H_BUF + idx] = ax;
                Hout[2 * H_BUF + idx] = axx;
                Hout[3 * H_BUF + idx] = axxx;
                Hout[4 * H_BUF + idx] = at;
            }
        }
        q ^= 1;
    }

    // ---- final linear layer (1x50) + moment accumulation ----
    float u = 0.0f, ux = 0.0f, uxxx = 0.0f, ut = 0.0f;
    if (hi == 0) {
        const float* Hin = H + q * 5 * H_BUF;
        u = b6[0];
        for (int k = 0; k < 50; ++k) {
            const float w = sm[OFF_W6 + k];
            const int idx = p * P_STRIDE + k;
            u    += w * Hin[0 * H_BUF + idx];
            ux   += w * Hin[1 * H_BUF + idx];
            uxxx += w * Hin[3 * H_BUF + idx];
            ut   += w * Hin[4 * H_BUF + idx];
        }
    }
    const float bs0 = u * ux, bs1 = uxxx, bs2 = ux;
    const float msk = (hi == 0 && valid) ? 1.0f : 0.0f;
    float v[10];
    v[0] = msk * ut * ut;
    v[1] = msk * ut * bs0;  v[2] = msk * ut * bs1;  v[3] = msk * ut * bs2;
    v[4] = msk * bs0 * bs0; v[5] = msk * bs0 * bs1; v[6] = msk * bs0 * bs2;
    v[7] = msk * bs1 * bs1; v[8] = msk * bs1 * bs2; v[9] = msk * bs2 * bs2;
#pragma unroll
    for (int k = 0; k < 10; ++k) {
        float t = v[k];
        t += __shfl_xor(t, 8, 32);
        t += __shfl_xor(t, 4, 32);
        t += __shfl_xor(t, 2, 32);
        t += __shfl_xor(t, 1, 32);
        if (lane == 0) atomicAdd(&acc[k], t);
    }
}

// loss_i = mean_j (u_t_j + p_i . b_j)^2 = A + 2 p.v + p^T M p   (closed form from 10 moments)
__global__ void para_kernel(const float* __restrict__ para,
                            const float* __restrict__ acc,
                            float* __restrict__ out)
{
    const int i = blockIdx.x * blockDim.x + threadIdx.x;
    if (i >= NPARA) return;
    const float p0 = para[3 * i], p1 = para[3 * i + 1], p2 = para[3 * i + 2];
    const float S0  = acc[0];
    const float v0  = acc[1], v1 = acc[2], v2 = acc[3];
    const float M00 = acc[4], M01 = acc[5], M02 = acc[6];
    const float M11 = acc[7], M12 = acc[8], M22 = acc[9];
    float r = S0
            + 2.0f * (p0 * v0 + p1 * v1 + p2 * v2)
            + p0 * p0 * M00 + p1 * p1 * M11 + p2 * p2 * M22
            + 2.0f * (p0 * p1 * M01 + p0 * p2 * M02 + p1 * p2 * M12);
    out[i] = r * (1.0f / (float)NPTS);
}

extern "C" void kernel_launch(void* const* d_in, const int* in_sizes, int n_in,
                              void* d_out, int out_size, void* d_ws, size_t ws_size,
                              hipStream_t stream)
{
    const float* x    = (const float*)d_in[0];
    const float* para = (const float*)d_in[1];
    const float* W1 = (const float*)d_in[2];  const float* b1 = (const float*)d_in[3];
    const float* W2 = (const float*)d_in[4];  const float* b2 = (const float*)d_in[5];
    const float* W3 = (const float*)d_in[6];  const float* b3 = (const float*)d_in[7];
    const float* W4 = (const float*)d_in[8];  const float* b4 = (const float*)d_in[9];
    const float* W5 = (const float*)d_in[10]; const float* b5 = (const float*)d_in[11];
    const float* W6 = (const float*)d_in[12]; const float* b6 = (const float*)d_in[13];
    float* acc = (float*)d_ws;
    float* out = (float*)d_out;

    zero_acc_kernel<<<1, 32, 0, stream>>>(acc);

    const int nwg = (NPTS + PTS_PER_WG - 1) / PTS_PER_WG;   // 157
    pinn_point_kernel<<<nwg, WAVES_PER_WG * 32, SMEM_FLOATS * sizeof(float), stream>>>(
        x, W1, b1, W2, b2, W3, b3, W4, b4, W5, b5, W6, b6, acc);

    para_kernel<<<(NPARA + 127) / 128, 128, 0, stream>>>(para, acc, out);
}